// GatedGCNConv_48782238547970
// MI455X (gfx1250) — compile-verified
//
#include <hip/hip_runtime.h>
#include <hip/hip_bf16.h>
#include <math.h>

typedef float v2f __attribute__((ext_vector_type(2)));
typedef float v8f __attribute__((ext_vector_type(8)));

#define N_GRAPHS 2048
#define APG 100
#define BPG 110
#define D 128
#define N_ATOMS (N_GRAPHS * APG)   // 204800
#define N_BONDS (N_GRAPHS * BPG)   // 225280
#define WPAD 132                   // padded LDS stride (dwords): bank-conflict free b64 frags

#define WMMA_F32(a, b, c) __builtin_amdgcn_wmma_f32_16x16x4_f32( \
    false, (a), false, (b), (short)0, (c), false, false)

// ---------------------------------------------------------------------------
// Y[n,128] = X[n,128] @ W[128,128] + b[128]   (fp32 WMMA 16x16x4)
// W staged TRANSPOSED+padded in LDS: B-frag = one contiguous ds_load_b64.
// 256 threads = 8 waves; each wave owns a 16x128 output row block.
// ---------------------------------------------------------------------------
__global__ __launch_bounds__(256) void gemm128_wmma(
    const float* __restrict__ X, const float* __restrict__ W,
    const float* __restrict__ bias, float* __restrict__ Y, int n)
{
    __shared__ __align__(16) float sW[128 * WPAD];
    for (int i = threadIdx.x; i < 128 * 128; i += 256) {
        int k = i >> 7, c = i & 127;
        sW[c * WPAD + k] = W[i];            // sW[col][k]
    }
    __syncthreads();

    const int wave = threadIdx.x >> 5;
    const int lane = threadIdx.x & 31;
    const int row0 = (blockIdx.x * 8 + wave) * 16;
    const int m  = lane & 15;
    const int kh = (lane >> 4) << 1;        // 0 | 2

    v8f acc[8];
#pragma unroll
    for (int t = 0; t < 8; ++t) { v8f z = {}; acc[t] = z; }

    const float* xrow = X + (size_t)(row0 + m) * D;

    for (int k0 = 0; k0 < D; k0 += 4) {
        v2f a = *(const v2f*)(xrow + k0 + kh);     // A[m][k0+kh .. +1]
#pragma unroll
        for (int t = 0; t < 8; ++t) {
            v2f b = *(const v2f*)&sW[(t * 16 + m) * WPAD + k0 + kh]; // B[k][col]
            acc[t] = WMMA_F32(a, b, acc[t]);
        }
    }

    const int mhi = (lane >> 4) * 8;        // C/D: lane L, reg r -> M=r+8*(L/16), N=L%16
#pragma unroll
    for (int t = 0; t < 8; ++t) {
        float bv = bias[t * 16 + m];
#pragma unroll
        for (int r = 0; r < 8; ++r)
            Y[(size_t)(row0 + mhi + r) * D + t * 16 + m] = acc[t][r] + bv;
    }
}

// ---------------------------------------------------------------------------
// Dual-output GEMM: Y1 = X@W1+b1, Y2 = X@W2+b2 -- X read ONCE from HBM.
// 132 KB LDS (two transposed W's); 2 blocks/WGP on the 320 KB CDNA5 LDS pool.
// ---------------------------------------------------------------------------
__global__ __launch_bounds__(256) void gemm128x2_wmma(
    const float* __restrict__ X,
    const float* __restrict__ W1, const float* __restrict__ b1,
    const float* __restrict__ W2, const float* __restrict__ b2,
    float* __restrict__ Y1, float* __restrict__ Y2, int n)
{
    __shared__ __align__(16) float sW1[128 * WPAD];
    __shared__ __align__(16) float sW2[128 * WPAD];
    for (int i = threadIdx.x; i < 128 * 128; i += 256) {
        int k = i >> 7, c = i & 127;
        sW1[c * WPAD + k] = W1[i];
        sW2[c * WPAD + k] = W2[i];
    }
    __syncthreads();

    const int wave = threadIdx.x >> 5;
    const int lane = threadIdx.x & 31;
    const int row0 = (blockIdx.x * 8 + wave) * 16;
    const int m  = lane & 15;
    const int kh = (lane >> 4) << 1;

    v8f acc1[8], acc2[8];
#pragma unroll
    for (int t = 0; t < 8; ++t) { v8f z = {}; acc1[t] = z; acc2[t] = z; }

    const float* xrow = X + (size_t)(row0 + m) * D;

    for (int k0 = 0; k0 < D; k0 += 4) {
        v2f a = *(const v2f*)(xrow + k0 + kh);
#pragma unroll
        for (int t = 0; t < 8; ++t) {
            v2f b = *(const v2f*)&sW1[(t * 16 + m) * WPAD + k0 + kh];
            acc1[t] = WMMA_F32(a, b, acc1[t]);
        }
#pragma unroll
        for (int t = 0; t < 8; ++t) {
            v2f b = *(const v2f*)&sW2[(t * 16 + m) * WPAD + k0 + kh];
            acc2[t] = WMMA_F32(a, b, acc2[t]);
        }
    }

    const int mhi = (lane >> 4) * 8;
#pragma unroll
    for (int t = 0; t < 8; ++t) {
        float bv1 = b1[t * 16 + m];
        float bv2 = b2[t * 16 + m];
#pragma unroll
        for (int r = 0; r < 8; ++r) {
            size_t idx = (size_t)(row0 + mhi + r) * D + t * 16 + m;
            Y1[idx] = acc1[t][r] + bv1;
            Y2[idx] = acc2[t][r] + bv2;
        }
    }
}

// ---------------------------------------------------------------------------
// Be-GEMM with fused bond update (never materializes Be):
//   e_new = e + relu( e@WB + bB + Ah[a0] + Ah[a1] + Cu[g] )
// Ah (105 MB) is L2-resident (192 MB L2), so epilogue gathers are cheap.
// ---------------------------------------------------------------------------
__global__ __launch_bounds__(256) void gemm_bond_wmma(
    const float* __restrict__ e, const float* __restrict__ WB,
    const float* __restrict__ bB, const float* __restrict__ Ah,
    const float* __restrict__ Cu, const int* __restrict__ bond_atoms,
    const int* __restrict__ bond_graph, float* __restrict__ e_new)
{
    __shared__ __align__(16) float sW[128 * WPAD];
    for (int i = threadIdx.x; i < 128 * 128; i += 256) {
        int k = i >> 7, c = i & 127;
        sW[c * WPAD + k] = WB[i];
    }
    __syncthreads();

    const int wave = threadIdx.x >> 5;
    const int lane = threadIdx.x & 31;
    const int row0 = (blockIdx.x * 8 + wave) * 16;
    const int m  = lane & 15;
    const int kh = (lane >> 4) << 1;

    v8f acc[8];
#pragma unroll
    for (int t = 0; t < 8; ++t) { v8f z = {}; acc[t] = z; }

    const float* xrow = e + (size_t)(row0 + m) * D;

    for (int k0 = 0; k0 < D; k0 += 4) {
        v2f a = *(const v2f*)(xrow + k0 + kh);
#pragma unroll
        for (int t = 0; t < 8; ++t) {
            v2f b = *(const v2f*)&sW[(t * 16 + m) * WPAD + k0 + kh];
            acc[t] = WMMA_F32(a, b, acc[t]);
        }
    }

    float bv[8];
#pragma unroll
    for (int t = 0; t < 8; ++t) bv[t] = bB[t * 16 + m];

    const int mhi = (lane >> 4) * 8;
#pragma unroll
    for (int r = 0; r < 8; ++r) {
        const int mm = row0 + mhi + r;            // bond row
        const int a0 = bond_atoms[2 * mm];
        const int a1 = bond_atoms[2 * mm + 1];
        const int g  = bond_graph[mm];
        const float* Ah0 = Ah + (size_t)a0 * D;
        const float* Ah1 = Ah + (size_t)a1 * D;
        const float* Cug = Cu + (size_t)g * D;
        const float* erow = e + (size_t)mm * D;
        float* orow = e_new + (size_t)mm * D;
#pragma unroll
        for (int t = 0; t < 8; ++t) {
            int c = t * 16 + m;
            float pre = acc[t][r] + bv[t] + Ah0[c] + Ah1[c] + Cug[c];
            orow[c] = erow[c] + fmaxf(pre, 0.0f);
        }
    }
}

// ---------------------------------------------------------------------------
// Atom update: deterministic gather over the graph's 110 bonds (LDS-staged
// endpoints, no float atomics -> bitwise replay-deterministic).
// ---------------------------------------------------------------------------
__global__ __launch_bounds__(128) void atom_update(
    const float* __restrict__ e_new, const float* __restrict__ Eh,
    const float* __restrict__ Dh, const float* __restrict__ Fu,
    const float* __restrict__ h, const int* __restrict__ bond_atoms,
    const int* __restrict__ atom_graph, float* __restrict__ h_new)
{
    __shared__ int sba[2 * BPG];
    const int a = blockIdx.x;
    const int g = atom_graph[a];
    const int bbase = g * BPG;
    for (int i = threadIdx.x; i < 2 * BPG; i += 128)
        sba[i] = bond_atoms[(size_t)bbase * 2 + i];
    __syncthreads();

    const int d = threadIdx.x;
    float num = 0.0f, den = 0.0f;
    for (int bb = 0; bb < BPG; ++bb) {
        int a0 = sba[2 * bb], a1 = sba[2 * bb + 1];
        if (a0 == a || a1 == a) {
            float s = 1.0f / (1.0f + __expf(-e_new[(size_t)(bbase + bb) * D + d]));
            if (a0 == a) { num += s * Eh[(size_t)a1 * D + d]; den += s; }
            if (a1 == a) { num += s * Eh[(size_t)a0 * D + d]; den += s; }
        }
    }
    float hagg = num / (den + 1e-6f) + Fu[(size_t)g * D + d];
    float pre = hagg + Dh[(size_t)a * D + d];
    h_new[(size_t)a * D + d] = h[(size_t)a * D + d] + fmaxf(pre, 0.0f);
}

// ---------------------------------------------------------------------------
// Per-graph means (linear layers commute with the mean: mean-then-GEMM
// replaces two full-size GEMMs + 1.6 GB of traffic).
// ---------------------------------------------------------------------------
__global__ __launch_bounds__(128) void graph_mean(
    const float* __restrict__ h_new, const float* __restrict__ e_new,
    float* __restrict__ ha_mean, float* __restrict__ eb_mean)
{
    const int g = blockIdx.x, d = threadIdx.x;
    float sa = 0.0f;
    for (int i = 0; i < APG; ++i) sa += h_new[(size_t)(g * APG + i) * D + d];
    ha_mean[(size_t)g * D + d] = sa * (1.0f / (float)APG);
    float sb = 0.0f;
    for (int i = 0; i < BPG; ++i) sb += e_new[(size_t)(g * BPG + i) * D + d];
    eb_mean[(size_t)g * D + d] = sb * (1.0f / (float)BPG);
}

__global__ __launch_bounds__(256) void global_update(
    const float* __restrict__ u, const float* __restrict__ Gm,
    const float* __restrict__ Hm, const float* __restrict__ Iu,
    float* __restrict__ u_new)
{
    const int i = blockIdx.x * 256 + threadIdx.x;
    float pre = Gm[i] + Hm[i] + Iu[i];
    u_new[i] = u[i] + fmaxf(pre, 0.0f);
}

extern "C" void kernel_launch(void* const* d_in, const int* in_sizes, int n_in,
                              void* d_out, int out_size, void* d_ws, size_t ws_size,
                              hipStream_t stream) {
    const float* h = (const float*)d_in[0];
    const float* e = (const float*)d_in[1];
    const float* u = (const float*)d_in[2];
    const int* bond_atoms = (const int*)d_in[3];
    const int* atom_graph = (const int*)d_in[4];
    const int* bond_graph = (const int*)d_in[5];
    const float* WA = (const float*)d_in[6],  *bA = (const float*)d_in[7];
    const float* WB = (const float*)d_in[8],  *bB = (const float*)d_in[9];
    const float* WC = (const float*)d_in[10], *bC = (const float*)d_in[11];
    const float* WD = (const float*)d_in[12], *bD = (const float*)d_in[13];
    const float* WE = (const float*)d_in[14], *bE = (const float*)d_in[15];
    const float* WF = (const float*)d_in[16], *bF = (const float*)d_in[17];
    const float* WG = (const float*)d_in[18], *bG = (const float*)d_in[19];
    const float* WH = (const float*)d_in[20], *bH = (const float*)d_in[21];
    const float* WI = (const float*)d_in[22], *bI = (const float*)d_in[23];

    // Outputs (concatenated): h_new, e_new, u_new
    float* h_new = (float*)d_out;
    float* e_new = h_new + (size_t)N_ATOMS * D;
    float* u_new = e_new + (size_t)N_BONDS * D;

    // Workspace layout (~217 MB)
    float* ws = (float*)d_ws;
    float* Ah = ws;                                 // [N_ATOMS, D]  (reused as Dh)
    float* Eh = Ah + (size_t)N_ATOMS * D;           // [N_ATOMS, D]
    float* Cu = Eh + (size_t)N_ATOMS * D;           // [N_GRAPHS, D] ...
    float* Fu = Cu + (size_t)N_GRAPHS * D;
    float* Iu = Fu + (size_t)N_GRAPHS * D;
    float* ha_mean = Iu + (size_t)N_GRAPHS * D;
    float* eb_mean = ha_mean + (size_t)N_GRAPHS * D;
    float* Gm = eb_mean + (size_t)N_GRAPHS * D;
    float* Hm = Gm + (size_t)N_GRAPHS * D;

    const dim3 blk256(256), blk128(128);
    const dim3 ga(N_ATOMS / 128);   // 1600
    const dim3 gb(N_BONDS / 128);   // 1760
    const dim3 gu(N_GRAPHS / 128);  // 16

    // Phase 1: projections (h read once for Ah+Eh)
    gemm128x2_wmma<<<ga, blk256, 0, stream>>>(h, WA, bA, WE, bE, Ah, Eh, N_ATOMS);
    gemm128_wmma<<<gu, blk256, 0, stream>>>(u, WC, bC, Cu, N_GRAPHS);
    gemm128_wmma<<<gu, blk256, 0, stream>>>(u, WF, bF, Fu, N_GRAPHS);
    gemm128_wmma<<<gu, blk256, 0, stream>>>(u, WI, bI, Iu, N_GRAPHS);

    // Phase 2: Be-GEMM + fused bond update -> e_new (direct to output)
    gemm_bond_wmma<<<gb, blk256, 0, stream>>>(e, WB, bB, Ah, Cu,
                                              bond_atoms, bond_graph, e_new);

    // Phase 3: Dh (reuse Ah buffer, now dead), then atom update -> h_new
    gemm128_wmma<<<ga, blk256, 0, stream>>>(h, WD, bD, Ah, N_ATOMS);
    atom_update<<<dim3(N_ATOMS), blk128, 0, stream>>>(e_new, Eh, Ah, Fu, h,
                                                      bond_atoms, atom_graph, h_new);

    // Phase 4: per-graph means, tiny GEMMs, global update -> u_new
    graph_mean<<<dim3(N_GRAPHS), blk128, 0, stream>>>(h_new, e_new, ha_mean, eb_mean);
    gemm128_wmma<<<gu, blk256, 0, stream>>>(ha_mean, WG, bG, Gm, N_GRAPHS);
    gemm128_wmma<<<gu, blk256, 0, stream>>>(eb_mean, WH, bH, Hm, N_GRAPHS);
    global_update<<<dim3(N_GRAPHS * D / 256), blk256, 0, stream>>>(u, Gm, Hm, Iu, u_new);
}